// Qwen3_5GatedDeltaNet_77300821393721
// MI455X (gfx1250) — compile-verified
//
#include <hip/hip_runtime.h>
#include <hip/hip_bf16.h>
#include <math.h>

typedef __attribute__((ext_vector_type(2))) float v2f;
typedef __attribute__((ext_vector_type(8))) float v8f;

#define B_    2
#define T_    4096
#define H_    16
#define DK_   128
#define DV_   128
#define CH_   64
#define NCH_  (T_ / CH_)

#define DVP   32            // DV columns handled per block (4-way DV split)
#define NVS   (DV_ / DVP)   // 4 slices -> 128 blocks total

#define LDW 132   // padded stride for 128-wide LDS arrays
#define LDT 68    // padded stride for 64-wide scratch
#define LDV 36    // padded stride for 32-wide (DV-slice) arrays

#define STQ_SZ (CH_ * DK_)  // staging sizes (floats, packed)
#define STV_SZ (CH_ * DVP)

__device__ __forceinline__ v8f wmma4(v2f a, v2f b, v8f c) {
  // D = A(16x4,f32) * B(4x16,f32) + C(16x16,f32)
  return __builtin_amdgcn_wmma_f32_16x16x4_f32(false, a, false, b, (short)0, c,
                                               false, false);
}
__device__ __forceinline__ v2f ld2(const float* p) {   // 8B-aligned LDS pair
  return *(const v2f*)p;
}
__device__ __forceinline__ unsigned lds_addr32(const void* p) {
  // LDS aperture: generic addr[31:0] is the in-workgroup LDS byte address
  return (unsigned)(unsigned long long)p;
}
__device__ __forceinline__ void async_b128(unsigned ldsb, unsigned voff,
                                           const void* sbase) {
  asm volatile("global_load_async_to_lds_b128 %0, %1, %2"
               :: "v"(ldsb), "v"(voff), "s"(sbase) : "memory");
}
__device__ __forceinline__ void async_b32(unsigned ldsb, unsigned voff,
                                          const void* sbase) {
  asm volatile("global_load_async_to_lds_b32 %0, %1, %2"
               :: "v"(ldsb), "v"(voff), "s"(sbase) : "memory");
}

__global__ __launch_bounds__(256)
void gdn_fwd_kernel(const float* __restrict__ q_g, const float* __restrict__ k_g,
                    const float* __restrict__ v_g, const float* __restrict__ g_g,
                    const float* __restrict__ b_g, float* __restrict__ out_g,
                    float* __restrict__ fs_g) {
  extern __shared__ float sm[];
  float* S    = sm;                       // [128][LDV]  state slice (DVP cols)
  float* qn   = S    + DK_ * LDV;         // [64][LDW]   l2norm(q)*scale
  float* kn   = qn   + CH_ * LDW;         // [64][LDW]   l2norm(k)
  float* wkc  = kn   + CH_ * LDW;         // [64][LDW]   k_beta*e^g -> k_cumdecay
  float* wu   = wkc  + CH_ * LDW;         // [64][LDV]   vb -> v_intra -> v_new
  float* scrM = wu   + CH_ * LDV;         // [64][LDT]   strict-lower M
  float* scrA = scrM + CH_ * LDT;         // [64][LDT]   intra attn (tril)
  float* gtmp = scrA + CH_ * LDT;         // [64]
  float* gg   = gtmp + CH_;               // [64] cumsum of g
  float* bet  = gg   + CH_;               // [64] beta
  float* eg   = bet  + CH_;               // [64] exp(gg)
  float* edk  = eg   + CH_;               // [64] exp(gg[63]-gg)
  float* stq  = edk  + CH_;               // [2][64*128] async q staging
  float* stk  = stq  + 2 * STQ_SZ;        // [2][64*128] async k staging
  float* stv  = stk  + 2 * STQ_SZ;        // [2][64*32]  async v-slice staging

  const int tid  = threadIdx.x;
  const int lane = tid & 31;
  const int wave = tid >> 5;              // 0..7
  const int bh   = blockIdx.x >> 2;       // (b,h) pair
  const int dv0  = (blockIdx.x & 3) * DVP;
  const int b    = bh / H_;
  const int h    = bh % H_;

  const int ln   = lane & 15;             // n (and m for A) within tile
  const int kh   = (lane >> 4) * 2;       // half-wave K sub-offset
  const int rb   = (lane >> 4) * 8;       // C/D row sub-offset

  for (int i = tid; i < DK_ * LDV; i += 256) S[i] = 0.f;

  // ---- async prefetch: each wave issues 24 ops (8 rows x {q,k,v}) ----
  auto prefetch = [&](int ckn, int buf) {
    const size_t tbase = (size_t)(b * T_ + ckn * CH_) * H_ + h;
    const float* qb = q_g + tbase * DK_;
    const float* kb = k_g + tbase * DK_;
    const float* vb = v_g + tbase * DV_ + dv0;
    const int r0 = wave * 8;
    unsigned lq = lds_addr32(stq + buf * STQ_SZ + r0 * DK_) + lane * 16u;
    unsigned lk = lds_addr32(stk + buf * STQ_SZ + r0 * DK_) + lane * 16u;
    unsigned lv = lds_addr32(stv + buf * STV_SZ + r0 * DVP) + lane * 4u;
    unsigned gq = (unsigned)(r0 * H_ * DK_ * 4) + lane * 16u;
    unsigned gv = (unsigned)(r0 * H_ * DV_ * 4) + lane * 4u;
    for (int rr = 0; rr < 8; ++rr) {
      async_b128(lq, gq, qb);
      async_b128(lk, gq, kb);
      async_b32(lv, gv, vb);
      lq += DK_ * 4; lk += DK_ * 4; lv += DVP * 4;
      gq += H_ * DK_ * 4; gv += H_ * DV_ * 4;
    }
  };

  prefetch(0, 0);                          // prologue: chunk 0 -> buffer 0

  const float qscale = 0.08838834764831845f;  // DK^-0.5

  for (int ck = 0; ck < NCH_; ++ck) {
    const int t0  = ck * CH_;
    const int cur = ck & 1;

    // issue next chunk's loads (clamped on last iter; lands in other buffer),
    // then wait until only those 24 remain -> current chunk's data is ready
    const int ckn = (ck + 1 < NCH_) ? ck + 1 : ck;
    prefetch(ckn, (ck + 1) & 1);
    asm volatile("s_wait_asynccnt 24" ::: "memory");
    __syncthreads();

    // ---------- stage 0: normalize from staging, beta-scale ----------
    {
      const int row = tid >> 2;            // 0..63
      const int qr  = tid & 3;
      const int c0  = qr * 32;             // 32 q/k cols per thread
      const float* sq_ = stq + cur * STQ_SZ + row * DK_;
      const float* sk_ = stk + cur * STQ_SZ + row * DK_;
      float sq = 0.f, sk = 0.f;
      for (int c = 0; c < 32; ++c) {
        float xq = sq_[c0 + c];
        float xk = sk_[c0 + c];
        qn[row * LDW + c0 + c] = xq;
        kn[row * LDW + c0 + c] = xk;
        sq += xq * xq;
        sk += xk * xk;
      }
      sq += __shfl_xor(sq, 1); sq += __shfl_xor(sq, 2);
      sk += __shfl_xor(sk, 1); sk += __shfl_xor(sk, 2);
      const float iq = rsqrtf(sq + 1e-6f) * qscale;
      const float ik = rsqrtf(sk + 1e-6f);
      const size_t tb = (size_t)(b * T_ + t0 + row);
      const float bb = b_g[tb * H_ + h];
      for (int c = 0; c < 32; ++c) {
        qn[row * LDW + c0 + c] *= iq;
        kn[row * LDW + c0 + c] *= ik;
      }
      const float* sv_ = stv + cur * STV_SZ + row * DVP;
      for (int c = 0; c < 8; ++c)          // 8 v-slice cols per thread
        wu[row * LDV + qr * 8 + c] = sv_[qr * 8 + c] * bb;
      if (qr == 0) {
        gtmp[row] = g_g[tb * H_ + h];
        bet[row]  = bb;
      }
    }
    __syncthreads();
    if (tid == 0) {                        // within-chunk cumsum of log-decay
      float s = 0.f;
      for (int i = 0; i < CH_; ++i) { s += gtmp[i]; gg[i] = s; }
    }
    __syncthreads();
    if (tid < CH_) {
      eg[tid]  = expf(gg[tid]);
      edk[tid] = expf(gg[CH_ - 1] - gg[tid]);
    }
    {
      const int row = tid >> 2;
      const int c0  = (tid & 3) * 32;
      const float sc = bet[row] * expf(gg[row]);   // k_beta * e^g
      for (int c = 0; c < 32; ++c)
        wkc[row * LDW + c0 + c] = kn[row * LDW + c0 + c] * sc;
    }
    __syncthreads();

    // ---------- stage 1: QK^T (waves 0-3) and KK^T (waves 4-7) via WMMA ----
    {
      const bool doM = (wave >= 4);
      const float* X = doM ? kn : qn;
      const int w4 = wave & 3;
      for (int tt = 0; tt < 4; ++tt) {
        const int tile = w4 * 4 + tt;             // 0..15 over 4x4 tiles
        const int row0 = (tile >> 2) * 16;
        const int col0 = (tile & 3) * 16;
        const int m = row0 + ln, n = col0 + ln;
        v8f acc = {};
        for (int k0 = 0; k0 < DK_; k0 += 4) {
          v2f a  = ld2(X  + m * LDW + k0 + kh);
          v2f b2 = ld2(kn + n * LDW + k0 + kh);
          acc = wmma4(a, b2, acc);
        }
        for (int r = 0; r < 8; ++r) {
          const int i = row0 + r + rb;
          const int j = col0 + ln;
          if (doM) {   // strict-lower, beta_i * exp(g_i - g_j)
            scrM[i * LDT + j] =
                (i > j) ? bet[i] * acc[r] * expf(gg[i] - gg[j]) : 0.f;
          } else {     // tril incl diag
            scrA[i * LDT + j] =
                (i >= j) ? acc[r] * expf(gg[i] - gg[j]) : 0.f;
          }
        }
      }
    }
    __syncthreads();

    // ---------- stage 2: blocked (I+M)^{-1} applied to [wu | wkc] ----------
    for (int ib = 0; ib < 4; ++ib) {
      const int r0 = ib * 16;
      if (tid < 160) {                     // waves 0..4 (wave-uniform split)
        float* buf; int c, ld;
        if (tid < 32) { buf = wu;  c = tid;      ld = LDV; }
        else          { buf = wkc; c = tid - 32; ld = LDW; }
        for (int r = 1; r < 16; ++r) {
          float acc = buf[(r0 + r) * ld + c];
          for (int j = 0; j < r; ++j)
            acc -= scrM[(r0 + r) * LDT + r0 + j] * buf[(r0 + j) * ld + c];
          buf[(r0 + r) * ld + c] = acc;
        }
      }
      __syncthreads();
      if (ib < 3) {
        const int ntiles = (3 - ib) * 10;  // 2 wu-tiles + 8 wkc-tiles per row
        for (int t = wave; t < ntiles; t += 8) {
          const int i2 = ib + 1 + t / 10;
          const int tc = t % 10;
          float* buf; int ld, col0;
          if (tc < 2) { buf = wu;  ld = LDV; col0 = tc * 16; }
          else        { buf = wkc; ld = LDW; col0 = (tc - 2) * 16; }
          const int mrow = i2 * 16 + ln;
          const int n = col0 + ln;
          v8f acc = {};
          for (int k0 = 0; k0 < 16; k0 += 4) {
            v2f a = ld2(scrM + mrow * LDT + r0 + k0 + kh);
            v2f b2;
            b2.x = buf[(r0 + k0 + kh) * ld + n];
            b2.y = buf[(r0 + k0 + kh + 1) * ld + n];
            acc = wmma4(a, b2, acc);
          }
          for (int r = 0; r < 8; ++r)
            buf[(i2 * 16 + r + rb) * ld + n] -= acc[r];
        }
      }
      __syncthreads();
    }

    // ---------- stage 3: v_new = v_intra - k_cumdecay @ S ----------
    {
      const int row0 = (wave >> 1) * 16;   // 4 row-tiles x 2 col-tiles
      const int col0 = (wave & 1) * 16;
      const int m = row0 + ln, n = col0 + ln;
      v8f acc = {};
      for (int k0 = 0; k0 < DK_; k0 += 4) {
        v2f a = ld2(wkc + m * LDW + k0 + kh);
        v2f b2;
        b2.x = S[(k0 + kh) * LDV + n];
        b2.y = S[(k0 + kh + 1) * LDV + n];
        acc = wmma4(a, b2, acc);
      }
      for (int r = 0; r < 8; ++r) {
        const int i = row0 + r + rb;
        wu[i * LDV + n] -= acc[r];
      }
    }
    __syncthreads();

    // ---------- stage 4: out = (q*e^g) @ S + attn @ v_new ----------
    {
      const int row0 = (wave >> 1) * 16;
      const int col0 = (wave & 1) * 16;
      const int m = row0 + ln, n = col0 + ln;
      const float egm = eg[m];
      v8f acc = {};
      for (int k0 = 0; k0 < DK_; k0 += 4) {      // inter-chunk term
        v2f a = ld2(qn + m * LDW + k0 + kh);
        a.x *= egm; a.y *= egm;
        v2f b2;
        b2.x = S[(k0 + kh) * LDV + n];
        b2.y = S[(k0 + kh + 1) * LDV + n];
        acc = wmma4(a, b2, acc);
      }
      for (int k0 = 0; k0 < CH_; k0 += 4) {      // intra-chunk term
        v2f a = ld2(scrA + m * LDT + k0 + kh);
        v2f b2;
        b2.x = wu[(k0 + kh) * LDV + n];
        b2.y = wu[(k0 + kh + 1) * LDV + n];
        acc = wmma4(a, b2, acc);
      }
      for (int r = 0; r < 8; ++r) {
        const int i = row0 + r + rb;
        __builtin_nontemporal_store(
            acc[r],
            out_g + ((size_t)(b * T_ + t0 + i) * H_ + h) * DV_ + dv0 + n);
      }
    }
    __syncthreads();

    // ---------- stage 5: S = e^{g_last} S + (k*e^{g_last-g})^T @ v_new -----
    {
      const float egl = eg[CH_ - 1];
      for (int tt = 0; tt < 2; ++tt) {
        const int tile = wave * 2 + tt;         // 0..15 over 8(dk) x 2(dv)
        const int row0 = (tile >> 1) * 16;      // dk
        const int col0 = (tile & 1) * 16;       // dv
        const int m = row0 + ln, n = col0 + ln;
        v8f acc = {};
        for (int k0 = 0; k0 < CH_; k0 += 4) {
          const int k1 = k0 + kh;
          v2f a, b2;
          a.x  = kn[k1 * LDW + m] * edk[k1];          // transposed A read
          a.y  = kn[(k1 + 1) * LDW + m] * edk[k1 + 1];
          b2.x = wu[k1 * LDV + n];
          b2.y = wu[(k1 + 1) * LDV + n];
          acc = wmma4(a, b2, acc);
        }
        for (int r = 0; r < 8; ++r) {
          const int i = row0 + r + rb;
          S[i * LDV + n] = S[i * LDV + n] * egl + acc[r];
        }
      }
    }
    __syncthreads();
  }

  // drain the (redundant) last prefetch before the wave can exit
  asm volatile("s_wait_asynccnt 0" ::: "memory");

  // ---------- final state [B,H,DK,DV] (this block's DV slice) ----------
  for (int idx = tid; idx < DK_ * DVP; idx += 256) {
    const int dk = idx >> 5, dv = idx & (DVP - 1);
    __builtin_nontemporal_store(
        S[dk * LDV + dv],
        fs_g + ((size_t)(b * H_ + h) * DK_ + dk) * DV_ + dv0 + dv);
  }
}

extern "C" void kernel_launch(void* const* d_in, const int* in_sizes, int n_in,
                              void* d_out, int out_size, void* d_ws, size_t ws_size,
                              hipStream_t stream) {
  const float* q    = (const float*)d_in[0];
  const float* k    = (const float*)d_in[1];
  const float* v    = (const float*)d_in[2];
  const float* g    = (const float*)d_in[3];
  const float* beta = (const float*)d_in[4];
  float* out = (float*)d_out;
  float* fs  = out + (size_t)B_ * T_ * H_ * DV_;   // tuple part 2: final_state

  const size_t shmem =
      (size_t)(DK_ * LDV + 3 * CH_ * LDW + CH_ * LDV + 2 * CH_ * LDT + 5 * CH_ +
               2 * (2 * STQ_SZ) + 2 * STV_SZ) *
      sizeof(float);   // ~305 KB, within CDNA5's 320 KB/WGP

  hipFuncSetAttribute((const void*)gdn_fwd_kernel,
                      hipFuncAttributeMaxDynamicSharedMemorySize, (int)shmem);

  gdn_fwd_kernel<<<B_ * H_ * NVS, 256, shmem, stream>>>(q, k, v, g, beta, out, fs);
}